// TokenFlowBlock_45208825757852
// MI455X (gfx1250) — compile-verified
//
#include <hip/hip_runtime.h>
#include <cstddef>

// ---------------- problem constants ----------------
#define DIMD   1024
#define SEQL   1024
#define NHEADS 16
#define NKVH   4
#define HD     64
#define NBLK   128
#define BLKSZ  8
#define BATCH  4
#define MROWS  (BATCH*SEQL)   // 4096 token rows
#define VROWS  (BATCH*NBLK)   // 512 modulation rows
#define MODN   (3*DIMD)       // 3072

typedef float v2f __attribute__((ext_vector_type(2)));
typedef float v8f __attribute__((ext_vector_type(8)));

// CDNA5 fp32 matrix op: D(16x16) = A(16x4) * B(4x16) + C, full fp32 precision.
__device__ __forceinline__ v8f wmma_f32(v2f a, v2f b, v8f c) {
  return __builtin_amdgcn_wmma_f32_16x16x4_f32(false, a, false, b, (short)0, c,
                                               false, false);
}

enum { EPI_STORE = 0, EPI_BIAS = 1, EPI_ROPE = 2, EPI_BHSD = 3, EPI_GATERES = 4 };

// ------------------------------------------------------------------
// Generic  out = A(MxK) @ W(NxK)^T  with fused epilogues.
// Wave tile 64x64 (16 f32-WMMA accumulators), 2x2 waves -> 128x128 / WG.
// All dims here are multiples of 128 (M in {512,4096}, N in {256,1024,3072}, K=1024).
// ------------------------------------------------------------------
template <int EPI>
__global__ __launch_bounds__(128) void gemm_wmma(
    const float* __restrict__ A, const float* __restrict__ W,
    float* __restrict__ out, int M, int N, int K,
    const float* __restrict__ p0,  // bias | freqs_cos | residual
    const float* __restrict__ p1,  // freqs_sin | mod (gate at +2*DIM)
    int aux)                       // #heads for (B,H,S,D) relayout
{
  const int lane = threadIdx.x & 31;
  const int wave = threadIdx.x >> 5;
  const int lo   = lane & 15;       // m (A rows) / n (B cols)
  const int hi   = lane >> 4;       // K half-select
  const int kb   = hi * 2;
  const int tileN = blockIdx.x * 128 + (wave & 1) * 64;
  const int tileM = blockIdx.y * 128 + (wave >> 1) * 64;

  v8f c[4][4] = {};
  const float* aBase = A + (size_t)(tileM + lo) * K + kb;
  const float* bBase = W + (size_t)(tileN + lo) * K + kb;

  for (int k0 = 0; k0 < K; k0 += 4) {
    if (k0 + 64 < K) {  // gfx1250 global_prefetch_b8 into L2/L0
      __builtin_prefetch(aBase + k0 + 64, 0, 1);
      __builtin_prefetch(bBase + k0 + 64, 0, 1);
    }
    v2f af[4], bf[4];
#pragma unroll
    for (int i = 0; i < 4; ++i)
      af[i] = *(const v2f*)(aBase + (size_t)i * 16 * K + k0);
#pragma unroll
    for (int j = 0; j < 4; ++j)
      bf[j] = *(const v2f*)(bBase + (size_t)j * 16 * K + k0);
#pragma unroll
    for (int i = 0; i < 4; ++i)
#pragma unroll
      for (int j = 0; j < 4; ++j)
        c[i][j] = wmma_f32(af[i], bf[j], c[i][j]);
  }

  // Epilogue.  C layout: vgpr r -> row = r + 8*hi, col = lo (ISA 7.12.2).
#pragma unroll
  for (int i = 0; i < 4; ++i) {
#pragma unroll
    for (int j = 0; j < 4; ++j) {
      v8f acc = c[i][j];
#pragma unroll
      for (int r = 0; r < 8; ++r) {
        int row = tileM + i * 16 + r + hi * 8;
        int col = tileN + j * 16 + lo;
        float val = acc[r];
        if constexpr (EPI == EPI_STORE) {
          out[(size_t)row * N + col] = val;
        } else if constexpr (EPI == EPI_BIAS) {
          out[(size_t)row * N + col] = val + p0[col];
        } else if constexpr (EPI == EPI_GATERES) {
          int b = row >> 10, s = row & (SEQL - 1);
          float g = p1[(size_t)(b * NBLK + (s >> 3)) * MODN + 2 * DIMD + col];
          out[(size_t)row * N + col] = p0[(size_t)row * N + col] + g * val;
        } else {  // EPI_ROPE / EPI_BHSD : relayout to (B, H, S, D)
          int b = row >> 10, s = row & (SEQL - 1);
          int h = col >> 6, d = col & (HD - 1);
          if constexpr (EPI == EPI_ROPE) {
            // pair (2i,2i+1) lives in adjacent lanes -> swap via shfl_xor(1)
            float other = __shfl_xor(val, 1, 32);
            float cv = p0[s * (HD / 2) + (d >> 1)];
            float sv = p1[s * (HD / 2) + (d >> 1)];
            val = (d & 1) ? (other * sv + val * cv) : (val * cv - other * sv);
          }
          out[(((size_t)(b * aux + h)) * SEQL + s) * HD + d] = val;
        }
      }
    }
  }
}

// ------------------------------------------------------------------
// RMSNorm + AdaLN modulation:  out = rmsnorm(x,w)*(1+scale) + shift
// one 256-thread block per token row
// ------------------------------------------------------------------
__global__ __launch_bounds__(256) void norm_mod_kernel(
    const float* __restrict__ x, const float* __restrict__ w,
    const float* __restrict__ mod, float* __restrict__ out)
{
  int row = blockIdx.x;  // b*SEQ + s
  int b = row >> 10, s = row & (SEQL - 1);
  const float* xr = x + (size_t)row * DIMD;
  const float* mr = mod + (size_t)(b * NBLK + (s >> 3)) * MODN;
  float4 xv = ((const float4*)xr)[threadIdx.x];
  __shared__ float red[256];
  red[threadIdx.x] = xv.x * xv.x + xv.y * xv.y + xv.z * xv.z + xv.w * xv.w;
  __syncthreads();
  for (int st = 128; st > 0; st >>= 1) {
    if (threadIdx.x < st) red[threadIdx.x] += red[threadIdx.x + st];
    __syncthreads();
  }
  float rms = rsqrtf(red[0] * (1.0f / DIMD) + 1e-6f);
  int d = threadIdx.x * 4;
  float4 o;
  o.x = xv.x * rms * w[d + 0] * (1.f + mr[DIMD + d + 0]) + mr[d + 0];
  o.y = xv.y * rms * w[d + 1] * (1.f + mr[DIMD + d + 1]) + mr[d + 1];
  o.z = xv.z * rms * w[d + 2] * (1.f + mr[DIMD + d + 2]) + mr[d + 2];
  o.w = xv.w * rms * w[d + 3] * (1.f + mr[DIMD + d + 3]) + mr[d + 3];
  ((float4*)(out + (size_t)row * DIMD))[threadIdx.x] = o;
}

__global__ __launch_bounds__(256) void silu_kernel(const float* __restrict__ in,
                                                   float* __restrict__ out, int n) {
  int i = blockIdx.x * 256 + threadIdx.x;
  if (i < n) { float v = in[i]; out[i] = v / (1.f + __expf(-v)); }
}

__global__ __launch_bounds__(256) void silumul_kernel(float* __restrict__ u,
                                                      const float* __restrict__ g,
                                                      int n) {
  int i = blockIdx.x * 256 + threadIdx.x;
  if (i < n) { float v = u[i]; u[i] = (v / (1.f + __expf(-v))) * g[i]; }
}

// ------------------------------------------------------------------
// Block-causal GQA attention.  One wave per (batch,head)x(16 q-rows).
// Scores for the allowed prefix (block-causal => k < r0+16) are built with
// f32 WMMA, staged in a padded LDS slab (CDNA5: 320KB/WGP), softmaxed,
// then consumed as A-fragments for the P*V WMMAs.
// grid = (SEQ/16, 64/4), block = 128, dynamic LDS = 4*16*1032*4 = 264192 B.
// ------------------------------------------------------------------
#define SSTR 1032
__global__ __launch_bounds__(128) void attn_kernel(
    const float* __restrict__ q, const float* __restrict__ k,
    const float* __restrict__ v, float* __restrict__ o)
{
  extern __shared__ float sm[];
  const int lane = threadIdx.x & 31;
  const int wave = threadIdx.x >> 5;
  const int lo = lane & 15, hi = lane >> 4;
  const int kb = hi * 2;
  float* sc = sm + (size_t)wave * 16 * SSTR;

  const int r0 = blockIdx.x * 16;             // q-row tile
  const int bh = blockIdx.y * 4 + wave;       // (batch,head)
  const int b = bh >> 4, h = bh & 15, kvh = h >> 2;
  const float* qb = q + (((size_t)(b * NHEADS + h)) * SEQL + r0) * HD;
  const float* kbp = k + (((size_t)(b * NKVH + kvh)) * SEQL) * HD;
  const float* vbp = v + (((size_t)(b * NKVH + kvh)) * SEQL) * HD;
  const int kend = r0 + 16;                   // block-causal bound

  // Preload Q as A-fragments: qa[t] covers d = 4t..4t+3
  v2f qa[16];
#pragma unroll
  for (int t = 0; t < 16; ++t)
    qa[t] = *(const v2f*)(qb + (size_t)lo * HD + t * 4 + kb);

  // ---- S = Q K^T * scale, masked, into LDS ----
  for (int c0 = 0; c0 < kend; c0 += 16) {
    v8f s8 = {};
#pragma unroll
    for (int t = 0; t < 16; ++t) {
      v2f bf = *(const v2f*)(kbp + (size_t)(c0 + lo) * HD + t * 4 + kb);
      s8 = wmma_f32(qa[t], bf, s8);
    }
#pragma unroll
    for (int r = 0; r < 8; ++r) {
      int row = r + hi * 8;
      int col = c0 + lo;
      float val = s8[r] * 0.125f;  // 1/sqrt(64)
      if (((r0 + row) >> 3) < (col >> 3)) val = -1.0e9f;  // block-causal mask
      sc[row * SSTR + col] = val;
    }
  }
  __syncthreads();

  // ---- softmax: 2 lanes per row (even/odd columns), shfl_xor(16) combine ----
  {
    int myrow = lo;
    float m = -3.0e38f;
    for (int j = hi; j < kend; j += 2) m = fmaxf(m, sc[myrow * SSTR + j]);
    m = fmaxf(m, __shfl_xor(m, 16, 32));
    float sum = 0.f;
    for (int j = hi; j < kend; j += 2) sum += __expf(sc[myrow * SSTR + j] - m);
    sum += __shfl_xor(sum, 16, 32);
    float inv = 1.0f / sum;
    for (int j = hi; j < kend; j += 2)
      sc[myrow * SSTR + j] = __expf(sc[myrow * SSTR + j] - m) * inv;
  }
  __syncthreads();

  // ---- O = P V  (P tiles re-read from LDS in A-fragment layout) ----
  v8f oacc[4] = {};
  for (int c0 = 0; c0 < kend; c0 += 16) {
#pragma unroll
    for (int ks = 0; ks < 4; ++ks) {
      v2f pa = *(const v2f*)(&sc[lo * SSTR + c0 + ks * 4 + kb]);
      int kk = c0 + ks * 4 + kb;
#pragma unroll
      for (int g = 0; g < 4; ++g) {
        v2f bv;
        bv.x = vbp[(size_t)kk * HD + g * 16 + lo];
        bv.y = vbp[(size_t)(kk + 1) * HD + g * 16 + lo];
        oacc[g] = wmma_f32(pa, bv, oacc[g]);
      }
    }
  }

  // store O back in (B, S, H*D) row-major for the wo GEMM
  float* ob = o + ((size_t)(b * SEQL + r0)) * (NHEADS * HD) + h * HD;
#pragma unroll
  for (int g = 0; g < 4; ++g)
#pragma unroll
    for (int r = 0; r < 8; ++r)
      ob[(size_t)(r + hi * 8) * (NHEADS * HD) + g * 16 + lo] = oacc[g][r];
}

// ------------------------------------------------------------------
// Host orchestration
// ------------------------------------------------------------------
extern "C" void kernel_launch(void* const* d_in, const int* in_sizes, int n_in,
                              void* d_out, int out_size, void* d_ws, size_t ws_size,
                              hipStream_t stream) {
  const float* x      = (const float*)d_in[0];
  const float* vec    = (const float*)d_in[1];
  const float* wq     = (const float*)d_in[2];
  const float* wk     = (const float*)d_in[3];
  const float* wv     = (const float*)d_in[4];
  const float* wo     = (const float*)d_in[5];
  const float* w1     = (const float*)d_in[6];
  const float* w2     = (const float*)d_in[7];
  const float* w3     = (const float*)d_in[8];
  const float* maw    = (const float*)d_in[9];
  const float* mab    = (const float*)d_in[10];
  const float* mfw    = (const float*)d_in[11];
  const float* mfb    = (const float*)d_in[12];
  const float* n1w    = (const float*)d_in[13];
  const float* n2w    = (const float*)d_in[14];
  const float* fcos   = (const float*)d_in[15];
  const float* fsin   = (const float*)d_in[16];
  float* out = (float*)d_out;

  // workspace layout (floats)
  const size_t KVSZ = (size_t)BATCH * NKVH * SEQL * HD;   // 1M floats (k or v)
  float* ws   = (float*)d_ws;
  float* svec = ws;                                   // 512*1024
  float* modA = svec + (size_t)VROWS * DIMD;          // 512*3072
  float* modF = modA + (size_t)VROWS * MODN;          // 512*3072
  float* bufD = modF + (size_t)VROWS * MODN;          // 4M  (xn -> attno -> w3 out)
  float* bufE = bufD + (size_t)MROWS * DIMD;          // 4M  (q -> hn)
  float* bufF = bufE + (size_t)MROWS * DIMD;          // 1M  (k)
  float* bufG = bufF + KVSZ;                          // 1M  (v)
  float* bufH = bufG + KVSZ;                          // 4M  (h)
  float* bufI = bufH + (size_t)MROWS * DIMD;          // 4M  (u = w1 out, then ffn)

  const dim3 blk128(128), blk256(256);

  // 1) svec = silu(vec)
  silu_kernel<<<(VROWS * DIMD + 255) / 256, blk256, 0, stream>>>(vec, svec, VROWS * DIMD);
  // 2,3) modulation GEMMs: (512 x 3072) = svec @ mod_w^T + b
  gemm_wmma<EPI_BIAS><<<dim3(MODN / 128, VROWS / 128), blk128, 0, stream>>>(
      svec, maw, modA, VROWS, MODN, DIMD, mab, nullptr, 0);
  gemm_wmma<EPI_BIAS><<<dim3(MODN / 128, VROWS / 128), blk128, 0, stream>>>(
      svec, mfw, modF, VROWS, MODN, DIMD, mfb, nullptr, 0);
  // 4) xn = rmsnorm(x)*(1+scale)+shift
  norm_mod_kernel<<<MROWS, blk256, 0, stream>>>(x, n1w, modA, bufD);
  // 5,6,7) Q/K/V projections (+RoPE for Q,K), relayout to (B,H,S,D)
  gemm_wmma<EPI_ROPE><<<dim3(DIMD / 128, MROWS / 128), blk128, 0, stream>>>(
      bufD, wq, bufE, MROWS, DIMD, DIMD, fcos, fsin, NHEADS);
  gemm_wmma<EPI_ROPE><<<dim3((NKVH * HD) / 128, MROWS / 128), blk128, 0, stream>>>(
      bufD, wk, bufF, MROWS, NKVH * HD, DIMD, fcos, fsin, NKVH);
  gemm_wmma<EPI_BHSD><<<dim3((NKVH * HD) / 128, MROWS / 128), blk128, 0, stream>>>(
      bufD, wv, bufG, MROWS, NKVH * HD, DIMD, nullptr, nullptr, NKVH);
  // 8) attention -> bufD (reuse, xn dead)
  const int attn_smem = 4 * 16 * SSTR * (int)sizeof(float);  // 264192 B (<=320KB/WGP)
  hipFuncSetAttribute((const void*)attn_kernel,
                      hipFuncAttributeMaxDynamicSharedMemorySize, attn_smem);
  attn_kernel<<<dim3(SEQL / 16, (BATCH * NHEADS) / 4), blk128, attn_smem, stream>>>(
      bufE, bufF, bufG, bufD);
  // 9) h = x + gate_a * (attno @ wo^T)
  gemm_wmma<EPI_GATERES><<<dim3(DIMD / 128, MROWS / 128), blk128, 0, stream>>>(
      bufD, wo, bufH, MROWS, DIMD, DIMD, x, modA, 0);
  // 10) hn = rmsnorm(h)*(1+scale_f)+shift_f  -> bufE (q dead)
  norm_mod_kernel<<<MROWS, blk256, 0, stream>>>(bufH, n2w, modF, bufE);
  // 11,12) u = hn @ w1^T -> bufI ; g = hn @ w3^T -> bufD
  gemm_wmma<EPI_STORE><<<dim3(DIMD / 128, MROWS / 128), blk128, 0, stream>>>(
      bufE, w1, bufI, MROWS, DIMD, DIMD, nullptr, nullptr, 0);
  gemm_wmma<EPI_STORE><<<dim3(DIMD / 128, MROWS / 128), blk128, 0, stream>>>(
      bufE, w3, bufD, MROWS, DIMD, DIMD, nullptr, nullptr, 0);
  // 13) u = silu(u) * g
  silumul_kernel<<<(MROWS * DIMD + 255) / 256, blk256, 0, stream>>>(bufI, bufD, MROWS * DIMD);
  // 14) out = h + gate_f * (u @ w2^T)
  gemm_wmma<EPI_GATERES><<<dim3(DIMD / 128, MROWS / 128), blk128, 0, stream>>>(
      bufI, w2, out, MROWS, DIMD, DIMD, bufH, modF, 0);
}